// CoRACommonEmbedding_64458869178500
// MI455X (gfx1250) — compile-verified
//
#include <hip/hip_runtime.h>

typedef __attribute__((ext_vector_type(2))) float v2f;
typedef __attribute__((ext_vector_type(8))) float v8f;

#define EMB_D 128
#define RANKS 8
#define SCALING_F 2.0f          // LORA_ALPHA / R = 16 / 8
#define RANK_THRESHOLD_F 0.1f

// One wave32 processes 16 tokens per iteration:
//   delta[16 tok x 128 dim] = loraB[tok][0:8] x effA[0:8][0:128]   (rank-8, fp32)
// computed as 8 N-tiles of V_WMMA_F32_16X16X4_F32, K=8 split into two K=4 steps.
__global__ __launch_bounds__(256) void cora_embed_wmma_kernel(
    const int* __restrict__ x,          // [N] token ids
    const float* __restrict__ emb,      // [VOCAB, 128]
    const float* __restrict__ loraA,    // [8, 128]
    const float* __restrict__ loraB,    // [VOCAB, 8]
    const float* __restrict__ rankPat,  // [8]
    float* __restrict__ out,            // [N, 128]
    int numGroups)                      // N / 16
{
    const int lane        = threadIdx.x & 31;
    const int waveInBlk   = threadIdx.x >> 5;
    const int wavesPerBlk = blockDim.x >> 5;
    const int gwave       = blockIdx.x * wavesPerBlk + waveInBlk;
    const int nwaves      = gridDim.x * wavesPerBlk;

    const int col  = lane & 15;   // N-index within a 16-wide tile
    const int half = lane >> 4;   // half-wave selects K pair (A) / K row (B)

    // AdaLoRA gate folded with SCALING: effA = loraA * (rank_pattern>thr) * SCALING
    float m[RANKS];
#pragma unroll
    for (int r = 0; r < RANKS; ++r)
        m[r] = (rankPat[r] > RANK_THRESHOLD_F) ? SCALING_F : 0.0f;

    // Precompute wave-resident B tiles (4x16 f32 per K-step, per N-tile).
    // B VGPR layout: v0 = row K0 (lanes 0-15) / K1 (lanes 16-31); v1 = K2/K3.
    v2f B1[8], B2[8];
#pragma unroll
    for (int t = 0; t < 8; ++t) {
        const int c = t * 16 + col;
        B1[t].x = loraA[(0 + half) * EMB_D + c] * m[0 + half];
        B1[t].y = loraA[(2 + half) * EMB_D + c] * m[2 + half];
        B2[t].x = loraA[(4 + half) * EMB_D + c] * m[4 + half];
        B2[t].y = loraA[(6 + half) * EMB_D + c] * m[6 + half];
    }

    for (int g = gwave; g < numGroups; g += nwaves) {
        const int tokBase = g * 16;
        // A-matrix: lane (L&15) owns token row; half selects K pair {0,1}/{2,3}.
        const int t_a = x[tokBase + col];
        const float* rowB = loraB + (size_t)t_a * RANKS;
        const v2f A1 = *reinterpret_cast<const v2f*>(rowB + half * 2);      // K 0..3
        const v2f A2 = *reinterpret_cast<const v2f*>(rowB + half * 2 + 4);  // K 4..7

#pragma unroll
        for (int t = 0; t < 8; ++t) {
            v8f acc = {};
            acc = __builtin_amdgcn_wmma_f32_16x16x4_f32(
                false, A1, false, B1[t], (short)0, acc, false, false);
            acc = __builtin_amdgcn_wmma_f32_16x16x4_f32(
                false, A2, false, B2[t], (short)0, acc, false, false);

            // D layout: acc[j] @ lane L -> token (half*8 + j), dim t*16 + (L&15).
            // Fused gather of emb row + add + store (64B coalesced per half-wave).
#pragma unroll
            for (int j = 0; j < 8; ++j) {
                const int tok = __shfl(t_a, half * 8 + j, 32);
                const size_t eoff = (size_t)tok * EMB_D + t * 16 + col;
                const size_t ooff = (size_t)(tokBase + half * 8 + j) * EMB_D + t * 16 + col;
                out[ooff] = emb[eoff] + acc[j];
            }
        }
    }
}

// Scalar tail for N % 16 tokens (not taken for the reference shape 819200).
__global__ void cora_embed_tail_kernel(
    const int* __restrict__ x, const float* __restrict__ emb,
    const float* __restrict__ loraA, const float* __restrict__ loraB,
    const float* __restrict__ rankPat, float* __restrict__ out,
    int startTok, int nTok)
{
    const int i = blockIdx.x * blockDim.x + threadIdx.x;
    if (i >= nTok * EMB_D) return;
    const int tokIdx = startTok + i / EMB_D;
    const int d      = i % EMB_D;
    const int t      = x[tokIdx];
    float acc = 0.0f;
#pragma unroll
    for (int r = 0; r < RANKS; ++r) {
        const float mr = (rankPat[r] > RANK_THRESHOLD_F) ? SCALING_F : 0.0f;
        acc += loraB[(size_t)t * RANKS + r] * loraA[r * EMB_D + d] * mr;
    }
    out[(size_t)tokIdx * EMB_D + d] = emb[(size_t)t * EMB_D + d] + acc;
}

extern "C" void kernel_launch(void* const* d_in, const int* in_sizes, int n_in,
                              void* d_out, int out_size, void* d_ws, size_t ws_size,
                              hipStream_t stream) {
    const int*   x       = (const int*)d_in[0];
    const float* emb     = (const float*)d_in[1];
    const float* loraA   = (const float*)d_in[2];
    const float* loraB   = (const float*)d_in[3];
    const float* rankPat = (const float*)d_in[4];
    float*       out     = (float*)d_out;

    const int nTok   = in_sizes[0];          // B*L = 819200
    const int groups = nTok / 16;
    const int rem    = nTok % 16;

    if (groups > 0) {
        const int wavesPerBlk = 8;           // 256 threads
        int blocks = (groups + wavesPerBlk - 1) / wavesPerBlk;
        if (blocks > 2048) blocks = 2048;    // grid-stride; amortize B-tile setup
        cora_embed_wmma_kernel<<<blocks, 256, 0, stream>>>(
            x, emb, loraA, loraB, rankPat, out, groups);
    }
    if (rem > 0) {
        const int startTok = groups * 16;
        const int work = rem * EMB_D;
        cora_embed_tail_kernel<<<(work + 255) / 256, 256, 0, stream>>>(
            x, emb, loraA, loraB, rankPat, out, startTok, rem);
    }
}